// ProgressiveKernelApply_42709154791906
// MI455X (gfx1250) — compile-verified
//
#include <hip/hip_runtime.h>

// ProgressiveKernelApply update step (splat=False), MI455X / gfx1250.
// data    [4,3,512,512] f32
// kernels [4,121,512,512] f32
// sum_r   [4,3,512,512] f32
// sum_w   [4,1,512,512] f32
// max_w   [4,1,512,512] f32
// out = [ new_sum_r (4*3*HW) | new_sum_w (4*HW) | new_max (4*HW) ]

#define HD   512
#define HW   (512 * 512)
#define KK   11
#define TAPS 121
#define RAD  5
#define TILE 16
#define HALO 26   // TILE + 2*RAD

// CDNA5 async global->LDS path (probe via __has_builtin so host pass and
// older toolchains fall back to predicated vector loads).
#if defined(__HIP_DEVICE_COMPILE__)
#if __has_builtin(__builtin_amdgcn_global_load_async_to_lds_b32) && \
    __has_builtin(__builtin_amdgcn_s_wait_asynccnt)
#define USE_ASYNC 1
#endif
#endif
#ifndef USE_ASYNC
#define USE_ASYNC 0
#endif

#define AS_GLOBAL __attribute__((address_space(1)))
#define AS_LDS    __attribute__((address_space(3)))

__global__ __launch_bounds__(256)
void ProgressiveKernelApply_42709154791906_kernel(
    const float* __restrict__ data,
    const float* __restrict__ kernels,
    const float* __restrict__ sum_r,
    const float* __restrict__ sum_w,
    const float* __restrict__ max_w,
    float* __restrict__ out)
{
    // Halo tile, channel-interleaved with stride 4 so one ds_load_b128
    // fetches ch0/ch1/ch2 for a pixel. 26*26*4*4B = 10.8 KB LDS.
    __shared__ __align__(16) float s_data[HALO * HALO * 4];

    const int tid = threadIdx.x;
    const int b   = blockIdx.z;
    const int ty0 = blockIdx.y * TILE;
    const int tx0 = blockIdx.x * TILE;

    const float* dbase = data + (size_t)b * 3 * HW;

    // ---------------- stage zero-padded data halo into LDS ----------------
#if USE_ASYNC
    // Pre-zero (covers out-of-bounds border), then async-copy valid pixels.
    for (int i = tid; i < HALO * HALO * 4; i += 256) s_data[i] = 0.0f;
    __syncthreads();
    for (int i = tid; i < HALO * HALO; i += 256) {
        const int row = i / HALO, col = i % HALO;
        const int ys = ty0 - RAD + row;
        const int xs = tx0 - RAD + col;
        if ((unsigned)ys < HD && (unsigned)xs < HD) {
            const float* g = dbase + (size_t)ys * HD + xs;
            __builtin_amdgcn_global_load_async_to_lds_b32(
                (AS_GLOBAL int*)(g),          (AS_LDS int*)&s_data[i * 4 + 0], 0, 0);
            __builtin_amdgcn_global_load_async_to_lds_b32(
                (AS_GLOBAL int*)(g + HW),     (AS_LDS int*)&s_data[i * 4 + 1], 0, 0);
            __builtin_amdgcn_global_load_async_to_lds_b32(
                (AS_GLOBAL int*)(g + 2 * HW), (AS_LDS int*)&s_data[i * 4 + 2], 0, 0);
        }
    }
    __builtin_amdgcn_s_wait_asynccnt(0);
    __syncthreads();
#else
    for (int i = tid; i < HALO * HALO; i += 256) {
        const int row = i / HALO, col = i % HALO;
        const int ys = ty0 - RAD + row;
        const int xs = tx0 - RAD + col;
        const bool valid = ((unsigned)ys < HD) && ((unsigned)xs < HD);
        const float* g = dbase + (size_t)ys * HD + xs;
        s_data[i * 4 + 0] = valid ? g[0] : 0.0f;
        s_data[i * 4 + 1] = valid ? g[HW] : 0.0f;
        s_data[i * 4 + 2] = valid ? g[2 * HW] : 0.0f;
    }
    __syncthreads();
#endif

    // ---------------- per-thread pixel ----------------
    const int px = tid & (TILE - 1);
    const int py = tid >> 4;
    const int x  = tx0 + px;
    const int y  = ty0 + py;
    const size_t pix = (size_t)y * HD + x;

    const float* kp = kernels + (size_t)b * TAPS * HW + pix;

    // Pass 1: running max over 121 taps, seeded with max_w.
    // Tap-major layout -> lane-coalesced 128B global loads; the tile's
    // 124 KB re-read in pass 2 hits L2.
    const float mw = max_w[(size_t)b * HW + pix];
    float m = mw;
    {
        const float* p = kp;
        #pragma unroll 11
        for (int t = 0; t < TAPS; ++t) { m = fmaxf(m, *p); p += HW; }
    }

    // Rescale old accumulators.
    const float scaler = __expf(mw - m);
    float r0 = sum_r[((size_t)b * 3 + 0) * HW + pix] * scaler;
    float r1 = sum_r[((size_t)b * 3 + 1) * HW + pix] * scaler;
    float r2 = sum_r[((size_t)b * 3 + 2) * HW + pix] * scaler;
    float w  = sum_w[(size_t)b * HW + pix] * scaler;

    // Validity bit-masks (zero-padding contributes to neither sum_r nor sum_w).
    unsigned rowm = 0, colm = 0;
    #pragma unroll
    for (int d = 0; d < KK; ++d) {
        if ((unsigned)(y + d - RAD) < HD) rowm |= 1u << d;
        if ((unsigned)(x + d - RAD) < HD) colm |= 1u << d;
    }

    // Pass 2: exp-accumulate over the 11x11 window.
    const float4* sd = reinterpret_cast<const float4*>(s_data);
    const float* p = kp;
    for (int dy = 0; dy < KK; ++dy) {
        const float4* srow = sd + (py + dy) * HALO + px;
        const float vy = ((rowm >> dy) & 1u) ? 1.0f : 0.0f;
        #pragma unroll
        for (int dx = 0; dx < KK; ++dx) {
            const float e = __expf(*p - m);   // v_exp_f32
            p += HW;
            const float4 d4 = srow[dx];       // ds_load_b128: ch0,ch1,ch2
            const float vf = ((colm >> dx) & 1u) ? vy : 0.0f;
            w  = fmaf(e, vf,   w);
            r0 = fmaf(e, d4.x, r0);
            r1 = fmaf(e, d4.y, r1);
            r2 = fmaf(e, d4.z, r2);
        }
    }

    // ---------------- outputs ----------------
    float* out_r = out;
    float* out_w = out + (size_t)4 * 3 * HW;
    float* out_m = out_w + (size_t)4 * HW;
    out_r[((size_t)b * 3 + 0) * HW + pix] = r0;
    out_r[((size_t)b * 3 + 1) * HW + pix] = r1;
    out_r[((size_t)b * 3 + 2) * HW + pix] = r2;
    out_w[(size_t)b * HW + pix] = w;
    out_m[(size_t)b * HW + pix] = m;
}

extern "C" void kernel_launch(void* const* d_in, const int* in_sizes, int n_in,
                              void* d_out, int out_size, void* d_ws, size_t ws_size,
                              hipStream_t stream)
{
    const float* data    = (const float*)d_in[0];
    const float* kernels = (const float*)d_in[1];
    const float* sum_r   = (const float*)d_in[2];
    const float* sum_w   = (const float*)d_in[3];
    const float* max_w   = (const float*)d_in[4];
    float* out = (float*)d_out;

    dim3 grid(HD / TILE, HD / TILE, 4);   // 32 x 32 x 4 tiles
    dim3 block(256);                      // 8 wave32s, one pixel per thread
    ProgressiveKernelApply_42709154791906_kernel<<<grid, block, 0, stream>>>(
        data, kernels, sum_r, sum_w, max_w, out);
}